// QuantumCircuit_65481071402825
// MI455X (gfx1250) — compile-verified
//
#include <hip/hip_runtime.h>
#include <hip/hip_bf16.h>

// ---------------------------------------------------------------------------
// 16-qubit batched statevector simulator, fused into 16x16 complex GEMMs so
// every gate application runs on v_wmma_f32_16x16x4_f32.
//
// Layout: amplitude index idx, bit q of idx == qubit q. Groups of 4 qubits:
//   g0=bits0-3, g1=bits4-7, g2=bits8-11, g3=bits12-15.
// Per layer:
//   passA (tiles = contiguous 256 amps over bits0-7, per (batch, highbyte)):
//     GEMM M0 = C23*C12*C01*(Rot3x..xRot0) on g0 axis,
//     GEMM Pre1 = Rot7x..xRot4 on g1 axis,
//     then CNOT(3,4) + C45*C56*C67 as a pure store permutation.
//   passB (tiles = stride-256 columns over bits8-15, per (batch, lowbyte)):
//     GEMM Pre2 on g2 axis, GEMM Pre3 on g3 axis,
//     then CNOT(7,8) (cond. on bit7 of low byte, const per tile),
//     C(8,9..11) perm, CNOT(11,12) (cond. on bit11), C(12,13..15) perm,
//     fused into store addressing.  Last layer fuses the signed |amp|^2
//     reduction for <Z0> (sign = bit0 of low byte, const per tile).
//
// Roofline: state = 512 x 2^16 c64 = 256 MB. Gate-by-gate would be ~72 GB of
// HBM traffic; this schedule does init(256MB) + 8 in-place passes(4GB) with a
// fused final reduction => ~4.5 GB ~ 0.19 ms @ 23.3 TB/s, ~69 GFLOP of f32
// WMMA (memory bound, so f32 precision is free).
// ---------------------------------------------------------------------------

typedef float v2f __attribute__((ext_vector_type(2)));
typedef float v8f __attribute__((ext_vector_type(8)));

#if defined(__HIP_DEVICE_COMPILE__) && \
    !__has_builtin(__builtin_amdgcn_wmma_f32_16x16x4_f32)
#error "missing __builtin_amdgcn_wmma_f32_16x16x4_f32 for gfx1250"
#endif

__device__ __forceinline__ v8f wmma4(v2f a, v2f b, v8f c) {
#if defined(__HIP_DEVICE_COMPILE__)
  // (neg_a, A, neg_b, B, c_mod, C, reuse_a, reuse_b)
  return __builtin_amdgcn_wmma_f32_16x16x4_f32(false, a, false, b, (short)0, c,
                                               false, false);
#else
  (void)a; (void)b;
  return c;
#endif
}

// complex 16x16 matrix in WMMA A-operand layout (4 K-blocks of 16x4).
// ISA 16x4 f32 A layout: lanes0-15: v0=K0,v1=K1; lanes16-31: v0=K2,v1=K3.
struct AMat { v2f re[4]; v2f im[4]; v2f nim[4]; };
// B operand (4 K-blocks of 4x16): v0 holds K=4c+hi, v1 holds K=4c+2+hi.
struct CB   { v2f r[4];  v2f i[4]; };

__device__ __forceinline__ void loadA(const float* __restrict__ m3, AMat& A) {
  const int lane = threadIdx.x & 31;
  const int mrow = lane & 15;
  const int hi   = lane >> 4;
#pragma unroll
  for (int c = 0; c < 4; ++c) {
    const int off = mrow * 16 + 4 * c + 2 * hi;
    A.re[c]  = *(const v2f*)(m3 + off);
    A.im[c]  = *(const v2f*)(m3 + 256 + off);
    A.nim[c] = *(const v2f*)(m3 + 512 + off);
  }
}

// D = A_complex * B_complex  (Re += Ar*Br + (-Ai)*Bi ; Im += Ar*Bi + Ai*Br)
__device__ __forceinline__ void cgemm(const AMat& A, const CB& B, v8f& dr, v8f& di) {
#pragma unroll
  for (int c = 0; c < 4; ++c) {
    dr = wmma4(A.re[c],  B.r[c], dr);
    dr = wmma4(A.nim[c], B.i[c], dr);
    di = wmma4(A.re[c],  B.i[c], di);
    di = wmma4(A.im[c],  B.r[c], di);
  }
}

// value-destination permutation of the intra-group CNOT chain
// C(0,1) then C(1,2) then C(2,3) in local nibble coordinates.
__device__ __forceinline__ int pi4(int x) {
  x ^= (x & 1) << 1;
  x ^= (x & 2) << 1;
  x ^= (x & 4) << 1;
  return x;
}

// -------------------------- setup kernels ----------------------------------

// mats layout: [layer][group][plane re/im/-im][i*16+j], 4*4*3*256 floats
__global__ __launch_bounds__(256) void buildMats(const float* __restrict__ params,
                                                 float* __restrict__ mats) {
  const int layer = blockIdx.x;
  const int t = threadIdx.x;
  __shared__ float Ur[16][2][2];
  __shared__ float Ui[16][2][2];
  if (t < 16) {
    const int q = t;
    const float* p = params + (layer * 16 + q) * 3;
    const float phi = p[0], th = p[1], om = p[2];
    const float c = cosf(0.5f * th), s = sinf(0.5f * th);
    float sA, cA, sB, cB;
    sincosf(0.5f * (phi + om), &sA, &cA);
    sincosf(0.5f * (phi - om), &sB, &cB);
    // Rot = RZ(om) RY(th) RZ(phi)
    Ur[q][0][0] =  cA * c;  Ui[q][0][0] = -sA * c;
    Ur[q][0][1] = -cB * s;  Ui[q][0][1] = -sB * s;
    Ur[q][1][0] =  cB * s;  Ui[q][1][0] = -sB * s;
    Ur[q][1][1] =  cA * c;  Ui[q][1][1] =  sA * c;
  }
  __syncthreads();
  const int i = t >> 4, j = t & 15;
  for (int g = 0; g < 4; ++g) {
    int row = i;
    if (g == 0) {           // fold C23*C12*C01 row-perm into group0 matrix
      int r = i;
      if (r & 4) r ^= 8;    // c23
      if (r & 2) r ^= 4;    // c12
      if (r & 1) r ^= 2;    // c01
      row = r;
    }
    float re = 1.f, im = 0.f;
#pragma unroll
    for (int tt = 0; tt < 4; ++tt) {
      const int bi = (row >> tt) & 1, bj = (j >> tt) & 1;
      const float ar = Ur[4 * g + tt][bi][bj], ai = Ui[4 * g + tt][bi][bj];
      const float nr = re * ar - im * ai;
      const float ni = re * ai + im * ar;
      re = nr; im = ni;
    }
    float* base = mats + (size_t)((layer * 4 + g) * 3) * 256;
    base[t] = re; base[256 + t] = im; base[512 + t] = -im;
  }
}

// gtab[b][g][i] = prod_t (bit t of i ? sin(a/2) : cos(a/2)), 512*64 floats
__global__ __launch_bounds__(256) void buildGtab(const float* __restrict__ inputs,
                                                 float* __restrict__ gtab) {
  const int b = blockIdx.x * 256 + threadIdx.x;
  if (b >= 512) return;
  float c[16], s[16];
#pragma unroll
  for (int q = 0; q < 16; ++q) {
    const float a = inputs[b * 16 + q];
    c[q] = cosf(0.5f * a);
    s[q] = sinf(0.5f * a);
  }
  for (int g = 0; g < 4; ++g)
    for (int i = 0; i < 16; ++i) {
      float p = 1.f;
#pragma unroll
      for (int tt = 0; tt < 4; ++tt)
        p *= ((i >> tt) & 1) ? s[4 * g + tt] : c[4 * g + tt];
      gtab[b * 64 + g * 16 + i] = p;
    }
}

// product state init (RY feature map applied to |0..0> analytically)
__global__ __launch_bounds__(256) void initState(const float* __restrict__ gtab,
                                                 float* __restrict__ sre,
                                                 float* __restrict__ sim) {
  const size_t e = ((size_t)blockIdx.x * 256 + threadIdx.x) * 4;
  const int b = (int)(e >> 16);
  const int idx = (int)(e & 65535);
  const float* G = gtab + b * 64;
  const float gh = G[16 + ((idx >> 4) & 15)] * G[32 + ((idx >> 8) & 15)] *
                   G[48 + ((idx >> 12) & 15)];
  const int j0 = idx & 15;
  float4 r;
  r.x = G[j0 + 0] * gh; r.y = G[j0 + 1] * gh;
  r.z = G[j0 + 2] * gh; r.w = G[j0 + 3] * gh;
  *(float4*)(sre + e) = r;
  *(float4*)(sim + e) = make_float4(0.f, 0.f, 0.f, 0.f);
}

// -------------------------- layer passes -----------------------------------

__global__ __launch_bounds__(256) void passA(const float* __restrict__ mats,
                                             float* __restrict__ sre,
                                             float* __restrict__ sim,
                                             int layer) {
  __shared__ float smr[8][256];
  __shared__ float smi[8][256];
  const int wave = threadIdx.x >> 5;
  const int lane = threadIdx.x & 31;
  const int n = lane & 15;
  const int hi = lane >> 4;
  AMat A0, A1;
  loadA(mats + (size_t)((layer * 4 + 0) * 3) * 256, A0);
  loadA(mats + (size_t)((layer * 4 + 1) * 3) * 256, A1);
  float* lr = smr[wave];
  float* li = smi[wave];
#pragma unroll 1
  for (int t = 0; t < 4; ++t) {
    const int tile = (blockIdx.x * 8 + wave) * 4 + t;   // 131072 tiles
    const int b = tile >> 8;
    const int h = tile & 255;
    float* gr = sre + ((size_t)b << 16) + h * 256;
    float* gi = sim + ((size_t)b << 16) + h * 256;
    // coalesced tile load -> wave-private LDS (element (j0,j1) at j0+16*j1)
    *(float4*)&lr[lane * 4]       = ((const float4*)gr)[lane];
    *(float4*)&lr[128 + lane * 4] = ((const float4*)gr)[lane + 32];
    *(float4*)&li[lane * 4]       = ((const float4*)gi)[lane];
    *(float4*)&li[128 + lane * 4] = ((const float4*)gi)[lane + 32];
    // B1[k=j0, n=j1]
    CB B;
#pragma unroll
    for (int c = 0; c < 4; ++c) {
      B.r[c].x = lr[n * 16 + 4 * c + hi];
      B.r[c].y = lr[n * 16 + 4 * c + 2 + hi];
      B.i[c].x = li[n * 16 + 4 * c + hi];
      B.i[c].y = li[n * 16 + 4 * c + 2 + hi];
    }
    v8f yr = {}, yi = {};
    cgemm(A0, B, yr, yi);                 // group0 matrix
    // transpose via LDS: Y[m=j0', n=j1] -> y[j0' + 16*j1]
#pragma unroll
    for (int r = 0; r < 8; ++r) {
      const int m = r + 8 * hi;
      lr[m + 16 * n] = yr[r];
      li[m + 16 * n] = yi[r];
    }
    // B2[k=j1, n=j0'] = y[j0' + 16*j1]
#pragma unroll
    for (int c = 0; c < 4; ++c) {
      B.r[c].x = lr[n + 16 * (4 * c + hi)];
      B.r[c].y = lr[n + 16 * (4 * c + 2 + hi)];
      B.i[c].x = li[n + 16 * (4 * c + hi)];
      B.i[c].y = li[n + 16 * (4 * c + 2 + hi)];
    }
    v8f zr = {}, zi = {};
    cgemm(A1, B, zr, zi);                 // group1 rotations (Pre1)
    // store with CNOT(3,4) (ctrl=bit3 of j0) + C45*C56*C67 permutation
    const int flip = (n >> 3) & 1;
#pragma unroll
    for (int r = 0; r < 8; ++r) {
      const int m = r + 8 * hi;           // j1 before perms
      const int j1f = pi4(m ^ flip);
      gr[j1f * 16 + n] = zr[r];
      gi[j1f * 16 + n] = zi[r];
    }
  }
}

__global__ __launch_bounds__(256) void passB(const float* __restrict__ mats,
                                             float* __restrict__ sre,
                                             float* __restrict__ sim,
                                             float* __restrict__ partial,
                                             int layer, int last) {
  __shared__ float smr[4096];
  __shared__ float smi[4096];
  const int tid = threadIdx.x;
  const int b = blockIdx.x >> 4;          // 512 batches * 16 l-blocks
  const int l0 = (blockIdx.x & 15) * 16;
  float* gr = sre + ((size_t)b << 16) + l0;
  float* gi = sim + ((size_t)b << 16) + l0;
  const int krow = tid >> 2;
  const int ll4 = (tid & 3) * 4;
  // coalesced staged load: element (l, k=j2+16*j3) at sm[k*16 + (l-l0)]
#pragma unroll
  for (int it = 0; it < 4; ++it) {
    const int k = krow + 64 * it;
    *(float4*)&smr[k * 16 + ll4] = *(const float4*)(gr + (size_t)k * 256 + ll4);
    *(float4*)&smi[k * 16 + ll4] = *(const float4*)(gi + (size_t)k * 256 + ll4);
  }
  __syncthreads();
  const int wave = tid >> 5;
  const int lane = tid & 31;
  const int n = lane & 15;
  const int hi = lane >> 4;
  AMat A2, A3;
  loadA(mats + (size_t)((layer * 4 + 2) * 3) * 256, A2);
  loadA(mats + (size_t)((layer * 4 + 3) * 3) * 256, A3);
#pragma unroll 1
  for (int tt = 0; tt < 2; ++tt) {
    const int ll = wave * 2 + tt;         // wave-private LDS column
    const int l = l0 + ll;
    // B1[k=j2, n=j3]
    CB B;
#pragma unroll
    for (int c = 0; c < 4; ++c) {
      B.r[c].x = smr[((4 * c + hi)     + 16 * n) * 16 + ll];
      B.r[c].y = smr[((4 * c + 2 + hi) + 16 * n) * 16 + ll];
      B.i[c].x = smi[((4 * c + hi)     + 16 * n) * 16 + ll];
      B.i[c].y = smi[((4 * c + 2 + hi) + 16 * n) * 16 + ll];
    }
    v8f yr = {}, yi = {};
    cgemm(A2, B, yr, yi);                 // Pre2 on group2 axis
#pragma unroll
    for (int r = 0; r < 8; ++r) {
      const int m = r + 8 * hi;
      smr[(m + 16 * n) * 16 + ll] = yr[r];
      smi[(m + 16 * n) * 16 + ll] = yi[r];
    }
    // B2[k=j3, n=j2']
#pragma unroll
    for (int c = 0; c < 4; ++c) {
      B.r[c].x = smr[(n + 16 * (4 * c + hi))     * 16 + ll];
      B.r[c].y = smr[(n + 16 * (4 * c + 2 + hi)) * 16 + ll];
      B.i[c].x = smi[(n + 16 * (4 * c + hi))     * 16 + ll];
      B.i[c].y = smi[(n + 16 * (4 * c + 2 + hi)) * 16 + ll];
    }
    v8f zr = {}, zi = {};
    cgemm(A3, B, zr, zi);                 // Pre3 on group3 axis
    if (last) {
      // fused <Z0>: sign = bit0 of final index = bit0(l), const per tile
      float acc = 0.f;
#pragma unroll
      for (int r = 0; r < 8; ++r) acc += zr[r] * zr[r] + zi[r] * zi[r];
#pragma unroll
      for (int off = 16; off > 0; off >>= 1) acc += __shfl_xor(acc, off, 32);
      if (lane == 0) partial[b * 256 + l] = (l & 1) ? -acc : acc;
    } else {
      // CNOT(7,8) cond bit7(l); C(8,9..11) perm; CNOT(11,12); C(12,..15) perm
      const int b7 = (l >> 7) & 1;
      const int j2f = pi4(n ^ b7);
      const int fl2 = (j2f >> 3) & 1;
#pragma unroll
      for (int r = 0; r < 8; ++r) {
        const int m = r + 8 * hi;         // j3 before perms
        const int j3f = pi4(m ^ fl2);
        smr[(j2f + 16 * j3f) * 16 + ll] = zr[r];
        smi[(j2f + 16 * j3f) * 16 + ll] = zi[r];
      }
    }
  }
  if (!last) {
    __syncthreads();
#pragma unroll
    for (int it = 0; it < 4; ++it) {
      const int k = krow + 64 * it;
      *(float4*)(gr + (size_t)k * 256 + ll4) = *(const float4*)&smr[k * 16 + ll4];
      *(float4*)(gi + (size_t)k * 256 + ll4) = *(const float4*)&smi[k * 16 + ll4];
    }
  }
}

__global__ __launch_bounds__(256) void reduceOut(const float* __restrict__ partial,
                                                 float* __restrict__ out) {
  const int b = blockIdx.x * 256 + threadIdx.x;
  if (b >= 512) return;
  float s = 0.f;
  for (int l = 0; l < 256; ++l) s += partial[b * 256 + l];
  out[b] = s;
}

// -------------------------- launcher ---------------------------------------

extern "C" void kernel_launch(void* const* d_in, const int* in_sizes, int n_in,
                              void* d_out, int out_size, void* d_ws, size_t ws_size,
                              hipStream_t stream) {
  (void)in_sizes; (void)n_in; (void)out_size;
  const float* inputs = (const float*)d_in[0];   // (512,16) f32
  const float* params = (const float*)d_in[1];   // (4,16,3) f32
  float* out = (float*)d_out;                    // (512,) f32

  // workspace: mats(12288) | gtab(32768) | stateRe(2^25) | stateIm | partial
  const size_t NEEDED = (size_t)67284992 * sizeof(float);   // ~269 MB
  if (ws_size < NEEDED) return;
  float* w = (float*)d_ws;
  float* mats = w;
  float* gtab = w + 12288;
  float* sre  = w + 45056;
  float* sim  = sre + (size_t)33554432;
  float* partial = sim + (size_t)33554432;

  buildMats<<<4, 256, 0, stream>>>(params, mats);
  buildGtab<<<2, 256, 0, stream>>>(inputs, gtab);
  initState<<<32768, 256, 0, stream>>>(gtab, sre, sim);
  for (int layer = 0; layer < 4; ++layer) {
    passA<<<4096, 256, 0, stream>>>(mats, sre, sim, layer);
    passB<<<8192, 256, 0, stream>>>(mats, sre, sim, partial, layer,
                                    layer == 3 ? 1 : 0);
  }
  reduceOut<<<2, 256, 0, stream>>>(partial, out);
}